// StochasticTwoLayerRGCN_3822520893710
// MI455X (gfx1250) — compile-verified
//
#include <hip/hip_runtime.h>
#include <math.h>

#define NN 50000
#define EE 800000
#define RR 3
#define INF_ 128
#define HID 64
#define OUTF 64
#define HEADS 4
#define SLOPE 0.2f
#define C1 256   // HEADS*HID
#define C2 256   // HEADS*OUTF

typedef __attribute__((ext_vector_type(2))) float v2f;
typedef __attribute__((ext_vector_type(8))) float v8f;

// ---------------- order-preserving float<->uint for atomicMax ----------------
__device__ __forceinline__ unsigned f2key(float f) {
    unsigned u = __float_as_uint(f);
    return (u & 0x80000000u) ? ~u : (u | 0x80000000u);
}
__device__ __forceinline__ float key2f(unsigned u) {
    return __uint_as_float((u & 0x80000000u) ? (u & 0x7FFFFFFFu) : ~u);
}

// ---------------- fill ----------------
__global__ void fill_u32(unsigned* __restrict__ p, unsigned val, long long n) {
    long long t = (long long)blockIdx.x * blockDim.x + threadIdx.x;
    if (t < n) p[t] = val;
}

// ---------------- fp32 WMMA GEMM: C[nrows x 256] = A[nrows x K] * B[K x 256]
// one wave -> 16 rows x 64 cols (4 WMMA tiles sharing the A operand)
__global__ void wmma_gemm_f32(const float* __restrict__ A,
                              const float* __restrict__ B,
                              float* __restrict__ C,
                              int nrows, int K) {
    const int wave = (blockIdx.x * blockDim.x + threadIdx.x) >> 5;
    const int lane = threadIdx.x & 31;
    const int m_tiles = nrows >> 4;
    const int n_groups = C1 / 64;          // 4
    if (wave >= m_tiles * n_groups) return;
    const int m_tile  = wave / n_groups;
    const int n_group = wave % n_groups;

    const int lrow  = lane & 15;           // M (for A) / N (for B,C)
    const int khalf = (lane >> 4) << 1;    // 0 or 2

    v8f acc0 = {}, acc1 = {}, acc2 = {}, acc3 = {};
    const float* __restrict__ Arow = A + (size_t)(m_tile * 16 + lrow) * K;
    const int nb = n_group * 64;

    for (int k0 = 0; k0 < K; k0 += 4) {
        v2f a;
        a.x = Arow[k0 + khalf];
        a.y = Arow[k0 + khalf + 1];
        const float* __restrict__ B0 = B + (size_t)(k0 + khalf) * C1 + nb + lrow;
        const float* __restrict__ B1 = B0 + C1;
        v2f b0, b1, b2, b3;
        b0.x = B0[0];   b0.y = B1[0];
        b1.x = B0[16];  b1.y = B1[16];
        b2.x = B0[32];  b2.y = B1[32];
        b3.x = B0[48];  b3.y = B1[48];
        acc0 = __builtin_amdgcn_wmma_f32_16x16x4_f32(false, a, false, b0, (short)0, acc0, false, false);
        acc1 = __builtin_amdgcn_wmma_f32_16x16x4_f32(false, a, false, b1, (short)0, acc1, false, false);
        acc2 = __builtin_amdgcn_wmma_f32_16x16x4_f32(false, a, false, b2, (short)0, acc2, false, false);
        acc3 = __builtin_amdgcn_wmma_f32_16x16x4_f32(false, a, false, b3, (short)0, acc3, false, false);
    }

    // D layout: VGPR v -> row M = v + 8*(lane/16); col N = lane%16
    const int mrow0 = m_tile * 16 + ((lane >> 4) << 3);
    float* __restrict__ Cb = C + (size_t)mrow0 * C1 + nb + lrow;
#pragma unroll
    for (int v = 0; v < 8; ++v) {
        Cb[(size_t)v * C1 + 0]  = acc0[v];
        Cb[(size_t)v * C1 + 16] = acc1[v];
        Cb[(size_t)v * C1 + 32] = acc2[v];
        Cb[(size_t)v * C1 + 48] = acc3[v];
    }
}

// ---------------- el/er attention logits: one thread per (node, head) ----------------
__global__ void eler_kernel(const float* __restrict__ h,
                            const float* __restrict__ al,
                            const float* __restrict__ ar,
                            float* __restrict__ el, float* __restrict__ er, int n) {
    int tid = blockIdx.x * blockDim.x + threadIdx.x;
    if (tid >= n * HEADS) return;
    int node = tid >> 2, head = tid & 3;
    const float4* __restrict__ hp  = (const float4*)(h + (size_t)node * C1 + head * HID);
    const float4* __restrict__ alp = (const float4*)(al + head * HID);
    const float4* __restrict__ arp = (const float4*)(ar + head * HID);
    float sl = 0.f, sr = 0.f;
#pragma unroll
    for (int i = 0; i < HID / 4; ++i) {
        float4 hv = hp[i], a = alp[i], b = arp[i];
        sl += hv.x * a.x + hv.y * a.y + hv.z * a.z + hv.w * a.w;
        sr += hv.x * b.x + hv.y * b.y + hv.z * b.z + hv.w * b.w;
    }
    el[tid] = sl;
    er[tid] = sr;
}

// ---------------- edge pass A: e = leaky_relu(el[src]+er[dst]); seg-max via atomicMax ----------------
__global__ void edge_max_kernel(const int* __restrict__ src, const int* __restrict__ dst,
                                const float* __restrict__ el, const float* __restrict__ er,
                                float* __restrict__ e, unsigned* __restrict__ m_u, int ne) {
    int eidx = blockIdx.x * blockDim.x + threadIdx.x;
    if (eidx >= ne) return;
    int s = src[eidx], d = dst[eidx];
    float4 es = *(const float4*)(el + (size_t)s * 4);
    float4 ed = *(const float4*)(er + (size_t)d * 4);
    float v[4] = {es.x + ed.x, es.y + ed.y, es.z + ed.z, es.w + ed.w};
    float4 out;
    float* o = &out.x;
#pragma unroll
    for (int hh = 0; hh < 4; ++hh) {
        float t = v[hh];
        t = (t > 0.f) ? t : t * SLOPE;
        o[hh] = t;
        atomicMax(m_u + (size_t)d * 4 + hh, f2key(t));
    }
    *(float4*)(e + (size_t)eidx * 4) = out;
}

// ---------------- edge pass B: p = exp(e - m[dst]); seg-sum z ----------------
__global__ void edge_exp_kernel(const int* __restrict__ dst,
                                float* __restrict__ e, const unsigned* __restrict__ m_u,
                                float* __restrict__ z, int ne) {
    int eidx = blockIdx.x * blockDim.x + threadIdx.x;
    if (eidx >= ne) return;
    int d = dst[eidx];
    float4 ev = *(const float4*)(e + (size_t)eidx * 4);
    uint4 mu = *(const uint4*)(m_u + (size_t)d * 4);
    float4 p;
    p.x = __expf(ev.x - key2f(mu.x));
    p.y = __expf(ev.y - key2f(mu.y));
    p.z = __expf(ev.z - key2f(mu.z));
    p.w = __expf(ev.w - key2f(mu.w));
    *(float4*)(e + (size_t)eidx * 4) = p;
    atomicAdd(z + (size_t)d * 4 + 0, p.x);
    atomicAdd(z + (size_t)d * 4 + 1, p.y);
    atomicAdd(z + (size_t)d * 4 + 2, p.z);
    atomicAdd(z + (size_t)d * 4 + 3, p.w);
}

// ---------------- edge pass C: acc[dst] += h[src] * alpha (64 threads/edge, float4 each) ----------------
__global__ void edge_msg_kernel(const int* __restrict__ src, const int* __restrict__ dst,
                                const float* __restrict__ h, const float* __restrict__ p,
                                const float* __restrict__ z, float* __restrict__ acc, int ne) {
    long long t = (long long)blockIdx.x * blockDim.x + threadIdx.x;
    int eidx = (int)(t >> 6);
    int c4   = (int)(t & 63);
    if (eidx >= ne) return;
    int s = src[eidx], d = dst[eidx];
    int head = c4 >> 4;                          // (c4*4)/64
    float alpha = p[(size_t)eidx * 4 + head] / fmaxf(z[(size_t)d * 4 + head], 1e-9f);
    float4 hv = *(const float4*)(h + (size_t)s * C1 + c4 * 4);
    float* ap = acc + (size_t)d * C1 + c4 * 4;
    atomicAdd(ap + 0, hv.x * alpha);
    atomicAdd(ap + 1, hv.y * alpha);
    atomicAdd(ap + 2, hv.z * alpha);
    atomicAdd(ap + 3, hv.w * alpha);
}

// ---------------- finalize layer 1: head-mean of (acc + sum_r b1) ----------------
__global__ void finalize1_kernel(const float* __restrict__ acc, const float* __restrict__ b1,
                                 float* __restrict__ h1, int n) {
    int t = blockIdx.x * blockDim.x + threadIdx.x;
    if (t >= n * HID) return;
    int node = t >> 6, f = t & 63;
    float s = 0.f;
#pragma unroll
    for (int head = 0; head < HEADS; ++head) {
        float bias = 0.f;
#pragma unroll
        for (int r = 0; r < RR; ++r) bias += b1[r * C1 + head * HID + f];
        s += acc[(size_t)node * C1 + head * HID + f] + bias;
    }
    h1[t] = s * 0.25f;
}

// ---------------- finalize layer 2: out += sum_r b2 ----------------
__global__ void finalize2_kernel(float* __restrict__ out, const float* __restrict__ b2, long long n) {
    long long t = (long long)blockIdx.x * blockDim.x + threadIdx.x;
    if (t >= n) return;
    int cf = (int)(t & (C2 - 1));
    float bias = 0.f;
#pragma unroll
    for (int r = 0; r < RR; ++r) bias += b2[r * C2 + cf];
    out[t] += bias;
}

static inline int gsz(long long n, int b) { return (int)((n + b - 1) / b); }

extern "C" void kernel_launch(void* const* d_in, const int* in_sizes, int n_in,
                              void* d_out, int out_size, void* d_ws, size_t ws_size,
                              hipStream_t stream) {
    const float* x   = (const float*)d_in[0];
    const int* e1    = (const int*)d_in[1];    // [R,2,E]
    const int* e2    = (const int*)d_in[2];
    const float* W1  = (const float*)d_in[3];  // [R,128,256]
    const float* al1 = (const float*)d_in[4];  // [R,4,64]
    const float* ar1 = (const float*)d_in[5];
    const float* b1  = (const float*)d_in[6];  // [R,256]
    const float* W2  = (const float*)d_in[7];  // [R,64,256]
    const float* al2 = (const float*)d_in[8];
    const float* ar2 = (const float*)d_in[9];
    const float* b2  = (const float*)d_in[10];
    float* out = (float*)d_out;

    // workspace layout (floats)
    float* ws  = (float*)d_ws;
    float* h   = ws;                                   // N*256
    float* acc = h   + (size_t)NN * C1;                // N*256
    float* h1  = acc + (size_t)NN * C1;                // N*64
    float* el  = h1  + (size_t)NN * HID;               // N*4
    float* er  = el  + (size_t)NN * 4;                 // N*4
    float* e   = er  + (size_t)NN * 4;                 // E*4 (e -> p in place)
    unsigned* m_u = (unsigned*)(e + (size_t)EE * 4);   // N*4
    float* z   = (float*)(m_u + (size_t)NN * 4);       // N*4

    const int B = 256;
    const long long gemm_threads = (long long)(NN / 16) * 4 * 32;  // 3125 m-tiles * 4 n-groups * wave32

    // ---- layer 1 ----
    fill_u32<<<gsz((long long)NN * C1, B), B, 0, stream>>>((unsigned*)acc, 0u, (long long)NN * C1);
    for (int r = 0; r < RR; ++r) {
        const int* src = e1 + (size_t)r * 2 * EE;
        const int* dst = src + EE;
        wmma_gemm_f32<<<gsz(gemm_threads, B), B, 0, stream>>>(x, W1 + (size_t)r * INF_ * C1, h, NN, INF_);
        eler_kernel<<<gsz((long long)NN * HEADS, B), B, 0, stream>>>(h, al1 + r * C1, ar1 + r * C1, el, er, NN);
        fill_u32<<<gsz((long long)NN * 4, B), B, 0, stream>>>(m_u, 0u, (long long)NN * 4);
        fill_u32<<<gsz((long long)NN * 4, B), B, 0, stream>>>((unsigned*)z, 0u, (long long)NN * 4);
        edge_max_kernel<<<gsz(EE, B), B, 0, stream>>>(src, dst, el, er, e, m_u, EE);
        edge_exp_kernel<<<gsz(EE, B), B, 0, stream>>>(dst, e, m_u, z, EE);
        edge_msg_kernel<<<gsz((long long)EE * 64, B), B, 0, stream>>>(src, dst, h, e, z, acc, EE);
    }
    finalize1_kernel<<<gsz((long long)NN * HID, B), B, 0, stream>>>(acc, b1, h1, NN);

    // ---- layer 2 (accumulate straight into d_out) ----
    fill_u32<<<gsz((long long)NN * C2, B), B, 0, stream>>>((unsigned*)out, 0u, (long long)NN * C2);
    for (int r = 0; r < RR; ++r) {
        const int* src = e2 + (size_t)r * 2 * EE;
        const int* dst = src + EE;
        wmma_gemm_f32<<<gsz(gemm_threads, B), B, 0, stream>>>(h1, W2 + (size_t)r * HID * C2, h, NN, HID);
        eler_kernel<<<gsz((long long)NN * HEADS, B), B, 0, stream>>>(h, al2 + r * C2, ar2 + r * C2, el, er, NN);
        fill_u32<<<gsz((long long)NN * 4, B), B, 0, stream>>>(m_u, 0u, (long long)NN * 4);
        fill_u32<<<gsz((long long)NN * 4, B), B, 0, stream>>>((unsigned*)z, 0u, (long long)NN * 4);
        edge_max_kernel<<<gsz(EE, B), B, 0, stream>>>(src, dst, el, er, e, m_u, EE);
        edge_exp_kernel<<<gsz(EE, B), B, 0, stream>>>(dst, e, m_u, z, EE);
        edge_msg_kernel<<<gsz((long long)EE * 64, B), B, 0, stream>>>(src, dst, h, e, z, out, EE);
    }
    finalize2_kernel<<<gsz((long long)NN * C2, B), B, 0, stream>>>(out, b2, (long long)NN * C2);
}